// CustomRNN_12987981103489
// MI455X (gfx1250) — compile-verified
//
#include <hip/hip_runtime.h>
#include <stdint.h>

#define NFULL   2128          // NX + 2*PML
#define NVAR    2048
#define PML     40
#define DT      0.0005f
#define DT2     (DT * DT)
#define INV_DX2 0.01f         // 1/(10*10)

// Tile: 14 output rows x 56 output cols per 256-thread block.
// LDS stage: 16 rows x 64 floats (1-row / 4-col halo), exactly one
// global_load_async_to_lds_b128 per thread (lane (tx,ty) -> row ty, quad tx).
#define TILE_X     56
#define TILE_Y     14
#define LDS_STRIDE 68         // floats per LDS row (16B-aligned pad)

typedef __attribute__((ext_vector_type(4))) float f4;
// Builtin parameter pointee type (from clang diagnostic in compile probe):
typedef int v4i __attribute__((vector_size(16)));
typedef __attribute__((address_space(1))) v4i* gv4i_p;
typedef __attribute__((address_space(3))) v4i* lv4i_p;

// ---- CDNA5 async global -> LDS (ASYNCcnt path) -----------------------------

__device__ __forceinline__ void async_b128(float* lds_dst, const float* gsrc) {
#if __has_builtin(__builtin_amdgcn_global_load_async_to_lds_b128)
  __builtin_amdgcn_global_load_async_to_lds_b128(
      (gv4i_p)gsrc, (lv4i_p)lds_dst, /*imm offset=*/0, /*cpol=*/0);
#else
  unsigned lds_off = (unsigned)(uintptr_t)lds_dst;
  asm volatile("global_load_async_to_lds_b128 %0, %1, off"
               :: "v"(lds_off), "v"(gsrc)
               : "memory");
#endif
}

__device__ __forceinline__ void wait_async_all() {
#if __has_builtin(__builtin_amdgcn_s_wait_asynccnt)
  __builtin_amdgcn_s_wait_asynccnt(0);
#else
  asm volatile("s_wait_asynccnt 0" ::: "memory");
#endif
}

// ---- Kernel A: fused stencil + alpha + border zeroing ----------------------

__global__ __launch_bounds__(256)
void wave_step_kernel(const float* __restrict__ p1,
                      const float* __restrict__ p2,
                      const float* __restrict__ varray,
                      float* __restrict__ pout) {
  __shared__ float tile[16 * LDS_STRIDE];

  const int tx = threadIdx.x;         // 0..15
  const int ty = threadIdx.y;         // 0..15
  const int x0 = blockIdx.x * TILE_X; // 0, 56, ..., 2072 (38 blocks, exact)
  const int y0 = blockIdx.y * TILE_Y; // 0, 14, ..., 2114 (152 blocks, exact)

  // One async b128 per thread: LDS row ty (global y0-1+ty),
  // LDS quad tx (global cols x0-4+4*tx .. +3). Clamped edge slots are only
  // ever consumed by border outputs, which are forced to zero below.
  {
    int gy = y0 - 1 + ty;
    gy = gy < 0 ? 0 : (gy > NFULL - 1 ? NFULL - 1 : gy);
    int gx = x0 - 4 + tx * 4;
    gx = gx < 0 ? 0 : (gx > NFULL - 4 ? NFULL - 4 : gx);
    async_b128(&tile[ty * LDS_STRIDE + tx * 4],
               p2 + (size_t)gy * NFULL + gx);
  }
  wait_async_all();
  __syncthreads();

  if (tx < 14 && ty < 14) {
    const int y  = y0 + ty;
    const int c0 = x0 + tx * 4;
    const int r  = ty + 1;
    const int lc = tx * 4 + 4;
    const bool yin = (y >= 8) && (y <= NFULL - 9);

    const f4 p1v = *(const f4*)(p1 + (size_t)y * NFULL + c0);
    f4 outv;

#pragma unroll
    for (int k = 0; k < 4; ++k) {
      const int x = c0 + k;
      float val = 0.0f;
      if (yin && x >= 8 && x <= NFULL - 9) {
        const float c = tile[r * LDS_STRIDE + lc + k];
        const float w = tile[r * LDS_STRIDE + lc + k - 1];
        const float e = tile[r * LDS_STRIDE + lc + k + 1];
        const float n = tile[(r - 1) * LDS_STRIDE + lc + k];
        const float s = tile[(r + 1) * LDS_STRIDE + lc + k];
        int vy = y - PML; vy = vy < 0 ? 0 : (vy > NVAR - 1 ? NVAR - 1 : vy);
        int vx = x - PML; vx = vx < 0 ? 0 : (vx > NVAR - 1 ? NVAR - 1 : vx);
        const float v     = varray[(size_t)vy * NVAR + vx];
        const float alpha = v * v * DT2;
        const float lap   = (w + e + n + s - 4.0f * c) * INV_DX2;
        val = 2.0f * c - p1v[k] + alpha * lap;
      }
      outv[k] = val;
    }
    *(f4*)(pout + (size_t)y * NFULL + c0) = outv;
  }
}

// ---- Kernel B: point source ------------------------------------------------

__global__ void add_source_kernel(float* __restrict__ p,
                                  const float* __restrict__ sf,
                                  const int* __restrict__ xs,
                                  const int* __restrict__ ys,
                                  const int* __restrict__ t) {
  const int row = xs[0] + PML;
  const int col = ys[0] + PML;
  p[(size_t)row * NFULL + col] += sf[t[0]] * DT2;
}

// ---- Kernel C: res[i] = p[PML+i][PML+50] -----------------------------------

__global__ __launch_bounds__(256)
void extract_col_kernel(const float* __restrict__ p,
                        float* __restrict__ res) {
  const int i = blockIdx.x * blockDim.x + threadIdx.x;
  if (i < NVAR) {
    res[i] = p[(size_t)(PML + i) * NFULL + (PML + 50)];
  }
}

// ---- Launch ---------------------------------------------------------------

extern "C" void kernel_launch(void* const* d_in, const int* in_sizes, int n_in,
                              void* d_out, int out_size, void* d_ws, size_t ws_size,
                              hipStream_t stream) {
  const float* p1     = (const float*)d_in[0];
  const float* p2     = (const float*)d_in[1];
  const float* varray = (const float*)d_in[2];
  const float* sf     = (const float*)d_in[3];
  const int*   xs     = (const int*)d_in[4];
  const int*   ys     = (const int*)d_in[5];
  const int*   t      = (const int*)d_in[6];

  float* pout = (float*)d_out;                         // 2128*2128 floats
  float* res  = pout + (size_t)NFULL * NFULL;          // 2048 floats

  dim3 blockA(16, 16);
  dim3 gridA(NFULL / TILE_X, NFULL / TILE_Y);          // 38 x 152, exact
  wave_step_kernel<<<gridA, blockA, 0, stream>>>(p1, p2, varray, pout);

  add_source_kernel<<<1, 1, 0, stream>>>(pout, sf, xs, ys, t);

  extract_col_kernel<<<(NVAR + 255) / 256, 256, 0, stream>>>(pout, res);
}